// YOLOLoss_27255862461044
// MI455X (gfx1250) — compile-verified
//
#include <hip/hip_runtime.h>
#include <hip/hip_bf16.h>
#include <math.h>

// ---------------- problem constants (from reference setup) ----------------
#define BS    32
#define TT    50          // targets per batch
#define NA    3           // anchors
#define HH    52
#define WW    52
#define NCLS  80
#define CPA   85          // channels per anchor (5 + 80)
#define PLANE (HH*WW)     // 2704
#define NCELL (BS*NA*HH*WW)   // 259584
#define DBLK  256
#define NWAVE (DBLK/32)       // 8 waves per block (wave32)
#define DGRID (NCELL/DBLK)    // 1014 exactly

// scaled anchors: ANCHORS / (416/52) = ANCHORS / 8
__device__ __constant__ float c_aw[NA] = {1.25f, 2.0f, 4.125f};
__device__ __constant__ float c_ah[NA] = {1.625f, 3.75f, 2.875f};

#ifndef __has_builtin
#define __has_builtin(x) 0
#endif
#if __has_builtin(__builtin_amdgcn_global_load_async_to_lds_b32)
#define HAVE_ASYNC_LDS 1
#else
#define HAVE_ASYNC_LDS 0
#endif

__device__ __forceinline__ float sig_(float z) { return 1.0f / (1.0f + expf(-z)); }

// reference _bce with clip(log(p), -100): logf(0) = -inf -> fmaxf -> -100
__device__ __forceinline__ float bce_(float p, float t) {
  float lp  = fmaxf(logf(p),        -100.0f);
  float l1p = fmaxf(logf(1.0f - p), -100.0f);
  return -(t * lp + (1.0f - t) * l1p);
}

// ---------------- kernel 1: per-target sparse precompute ----------------
// rec layout: bit31 valid | [5:0] gi | [11:6] gj | [13:12] best_n |
//             [20:14] cls_id | [23:21] (iou[a] > 0.5) flags
__global__ void k_targets(const float* __restrict__ tgt,
                          int* __restrict__ recs,
                          float4* __restrict__ tvals)
{
  int idx = blockIdx.x * blockDim.x + threadIdx.x;
  if (idx >= BS * TT) return;
  const float* p = tgt + (size_t)idx * 5;
  float cls = p[0], cx = p[1], cy = p[2], w = p[3], h = p[4];
  float t1 = cx / 416.0f, t2 = cy / 416.0f, t3 = w / 416.0f, t4 = h / 416.0f;
  float s = cls + t1 + t2 + t3 + t4;           // tgt.sum(-1)
  if (!(s > 0.0f)) { recs[idx] = 0; tvals[idx] = make_float4(0,0,0,0); return; }

  float gx = t1 * (float)WW, gy = t2 * (float)HH;
  float gw = t3 * (float)WW, gh = t4 * (float)HH;
  int gi = (int)gx, gj = (int)gy;              // truncation == floor (non-neg)

  float area_g = (gw + 1.0f) * (gh + 1.0f);
  float iou[NA];
  #pragma unroll
  for (int a = 0; a < NA; ++a) {
    float aw = c_aw[a], ah = c_ah[a];
    float inter = fmaxf(fminf(gw, aw) + 1.0f, 0.0f) * fmaxf(fminf(gh, ah) + 1.0f, 0.0f);
    float area_a = (aw + 1.0f) * (ah + 1.0f);
    iou[a] = inter / (area_g + area_a - inter + 1e-16f);
  }
  int bn = 0; float best = iou[0];             // argmax, first-max-wins (jnp)
  if (iou[1] > best) { bn = 1; best = iou[1]; }
  if (iou[2] > best) { bn = 2; best = iou[2]; }

  int ib = 0;
  #pragma unroll
  for (int a = 0; a < NA; ++a) if (iou[a] > 0.5f) ib |= (1 << a);

  int cid = (int)cls;
  unsigned rec = (1u << 31) | (unsigned)(gi & 63) | ((unsigned)(gj & 63) << 6) |
                 ((unsigned)bn << 12) | ((unsigned)(cid & 127) << 14) |
                 ((unsigned)ib << 21);
  recs[idx] = (int)rec;

  float tx = gx - (float)gi, ty = gy - (float)gj;
  float tw = logf(gw / c_aw[bn] + 1e-16f);
  float th = logf(gh / c_ah[bn] + 1e-16f);
  tvals[idx] = make_float4(tx, ty, tw, th);
}

// ---------------- kernel 2: one thread per (b,a,j,i) cell ----------------
__global__ __launch_bounds__(DBLK)
void k_dense(const float* __restrict__ in,
             const int* __restrict__ recs,
             const float4* __restrict__ tvals,
             float* __restrict__ partials)
{
  int tid = blockIdx.x * blockDim.x + threadIdx.x;
  int i = tid % WW;
  int j = (tid / WW) % HH;
  int a = (tid / PLANE) % NA;
  int b = tid / (PLANE * NA);
  size_t base_in = ((size_t)(b * NA + a) * CPA) * PLANE + (size_t)j * WW + i;

#if HAVE_ASYNC_LDS
  // gfx1250 async global->LDS: kick off the conf load, overlap with target scan
  __shared__ float sconf[DBLK];
  __builtin_amdgcn_global_load_async_to_lds_b32(
      (int*)(in + base_in + (size_t)4 * PLANE),
      (int*)&sconf[threadIdx.x], 0, 0);
#else
  __builtin_prefetch(in + base_in + (size_t)4 * PLANE, 0, 0);  // global_prefetch_b8
#endif

  // scan this batch's 50 targets (32 KB total, L0/L2-resident, broadcast reads)
  int rbase = b * TT;
  bool masked = false, noobj = true;
  unsigned cb0 = 0, cb1 = 0, cb2 = 0;
  int twin = -1;
  for (int t = 0; t < TT; ++t) {
    int rec = recs[rbase + t];
    if (rec >= 0) continue;                    // bit31 clear -> invalid target
    int gi = rec & 63, gj = (rec >> 6) & 63;
    if (gi != i || gj != j) continue;
    if ((rec >> (21 + a)) & 1) noobj = false;  // IoU[a] > thr clears noobj
    int bn = (rec >> 12) & 3;
    if (bn == a) {                             // mask scatter hits this cell
      masked = true; twin = t;                 // ascending t -> last wins
      int cid = (rec >> 14) & 127;
      if (cid < 32) cb0 |= 1u << cid;
      else if (cid < 64) cb1 |= 1u << (cid - 32);
      else cb2 |= 1u << (cid - 64);            // tcls duplicates: union of classes
    }
  }

#if HAVE_ASYNC_LDS
#if __has_builtin(__builtin_amdgcn_s_wait_asynccnt)
  __builtin_amdgcn_s_wait_asynccnt(0);
#else
  asm volatile("s_wait_asynccnt 0" ::: "memory");
#endif
  float conf_logit = sconf[threadIdx.x];
#else
  float conf_logit = in[base_in + (size_t)4 * PLANE];
#endif
  float conf = sig_(conf_logit);

  float s_x = 0.f, s_y = 0.f, s_w = 0.f, s_h = 0.f;
  float s_cpos = 0.f, s_noobj = 0.f, s_cls = 0.f, cnt = 0.f;

  if (noobj) s_noobj = -fmaxf(logf(1.0f - conf), -100.0f);  // bce(conf,0)

  if (masked) {
    cnt = 1.0f;
    s_cpos = -fmaxf(logf(conf), -100.0f);                   // bce(conf,1)
    __builtin_prefetch(in + base_in, 0, 0);
    float4 tv = tvals[rbase + twin];
    float xl = in[base_in + (size_t)0 * PLANE];
    float yl = in[base_in + (size_t)1 * PLANE];
    float wl = in[base_in + (size_t)2 * PLANE];
    float hl = in[base_in + (size_t)3 * PLANE];
    s_x = bce_(sig_(xl), tv.x);
    s_y = bce_(sig_(yl), tv.y);
    float dw = wl - tv.z, dh = hl - tv.w;
    s_w = dw * dw; s_h = dh * dh;
    __builtin_prefetch(in + base_in + (size_t)5 * PLANE, 0, 0);
    for (int c = 0; c < NCLS; ++c) {
      float pl = in[base_in + (size_t)(5 + c) * PLANE];
      unsigned bit = (c < 32) ? ((cb0 >> c) & 1u)
                   : (c < 64) ? ((cb1 >> (c - 32)) & 1u)
                              : ((cb2 >> (c - 64)) & 1u);
      s_cls += bce_(sig_(pl), bit ? 1.0f : 0.0f);
    }
  }

  // wave32 cross-lane shuffle reduction (deterministic fixed tree), then one
  // tiny LDS hop for the 8 waves of the block. All 256 lanes are always
  // active (NCELL % DBLK == 0), so no inactive-lane hazards.
  float vals[8] = { s_x, s_y, s_w, s_h, s_cpos, s_noobj, s_cls, cnt };
  #pragma unroll
  for (int k = 0; k < 8; ++k) {
    float v = vals[k];
    #pragma unroll
    for (int off = 16; off > 0; off >>= 1) v += __shfl_down(v, off, 32);
    vals[k] = v;                               // lane 0 holds the wave sum
  }
  __shared__ float wsum[NWAVE][8];
  int lane = threadIdx.x & 31;
  int wid  = threadIdx.x >> 5;
  if (lane == 0) {
    #pragma unroll
    for (int k = 0; k < 8; ++k) wsum[wid][k] = vals[k];
  }
  __syncthreads();
  if (threadIdx.x < 8) {                       // thread k: fixed-order sum over waves
    float s = 0.f;
    #pragma unroll
    for (int w = 0; w < NWAVE; ++w) s += wsum[w][threadIdx.x];
    partials[(size_t)blockIdx.x * 8 + threadIdx.x] = s;
  }
}

// ---------------- kernel 3: deterministic final reduction ----------------
__global__ void k_final(const float* __restrict__ partials, float* __restrict__ out)
{
  __shared__ double red[256];
  __shared__ double totals[8];
  double acc[8] = {0,0,0,0,0,0,0,0};
  for (int r = threadIdx.x; r < DGRID; r += 256) {
    #pragma unroll
    for (int k = 0; k < 8; ++k) acc[k] += (double)partials[(size_t)r * 8 + k];
  }
  for (int k = 0; k < 8; ++k) {
    red[threadIdx.x] = acc[k];
    __syncthreads();
    for (int off = 128; off > 0; off >>= 1) {
      if ((int)threadIdx.x < off) red[threadIdx.x] += red[threadIdx.x + off];
      __syncthreads();
    }
    if (threadIdx.x == 0) totals[k] = red[0];
    __syncthreads();
  }
  if (threadIdx.x == 0) {
    double N = (double)NCELL;
    double loss_x = totals[0] / N, loss_y = totals[1] / N;
    double loss_w = totals[2] / N, loss_h = totals[3] / N;
    double loss_conf = totals[4] / N + 0.5 * (totals[5] / N);
    double n_pos = totals[7];
    double loss_cls = totals[6] / (n_pos * (double)NCLS);
    double loss = (loss_x + loss_y) * 2.5 + (loss_w + loss_h) * 2.5
                + loss_conf * 1.0 + loss_cls * 1.0;
    out[0] = (float)loss;
  }
}

// ---------------- launch ----------------
extern "C" void kernel_launch(void* const* d_in, const int* in_sizes, int n_in,
                              void* d_out, int out_size, void* d_ws, size_t ws_size,
                              hipStream_t stream)
{
  const float* input   = (const float*)d_in[0];   // (32, 255, 52, 52) f32
  const float* targets = (const float*)d_in[1];   // (32, 50, 5) f32
  (void)in_sizes; (void)n_in; (void)out_size; (void)ws_size;

  char* ws = (char*)d_ws;
  int*    recs     = (int*)ws;                    // 1600 * 4  B = 6400
  float4* tvals    = (float4*)(ws + 6400);        // 1600 * 16 B = 25600 (16B aligned)
  float*  partials = (float*)(ws + 32000);        // 1014 * 8 * 4 B = 32448

  k_targets<<<(BS * TT + 255) / 256, 256, 0, stream>>>(targets, recs, tvals);
  k_dense  <<<DGRID, DBLK, 0, stream>>>(input, recs, tvals, partials);
  k_final  <<<1, 256, 0, stream>>>(partials, (float*)d_out);
}